// LmHeadAll_52201032516344
// MI455X (gfx1250) — compile-verified
//
#include <hip/hip_runtime.h>
#include <hip/hip_bf16.h>
#include <math.h>

// Problem constants (match reference)
#define BATCH   32
#define HIST    2048
#define HDIM    2048
#define VOCAB   128000
#define TOPK    50
#define MINKEEP 5
#define TOPP    0.8f
#define PEN     1.1f
#define LNEPS   1e-5f

typedef float v2f __attribute__((ext_vector_type(2)));
typedef float v8f __attribute__((ext_vector_type(8)));

// ---------------------------------------------------------------------------
// 0) zero the mark bitmap (graph-capture safe, no memset API needed)
// ---------------------------------------------------------------------------
__global__ void __launch_bounds__(256) zero_marks_kernel(uint4* __restrict__ marks, int n16) {
    int i = blockIdx.x * 256 + threadIdx.x;
    if (i < n16) marks[i] = make_uint4(0u, 0u, 0u, 0u);
}

// ---------------------------------------------------------------------------
// 1) LayerNorm over hidden dim: one block per batch row
// ---------------------------------------------------------------------------
__global__ void __launch_bounds__(256) layernorm_kernel(
    const float* __restrict__ x, const float* __restrict__ gamma,
    const float* __restrict__ beta, float* __restrict__ out) {
    __shared__ float s1[256], s2[256];
    const int row = blockIdx.x, tid = threadIdx.x;
    const float* xr = x + (size_t)row * HDIM;
    float sum = 0.f, sq = 0.f;
    for (int i = tid; i < HDIM; i += 256) { float v = xr[i]; sum += v; sq += v * v; }
    s1[tid] = sum; s2[tid] = sq; __syncthreads();
    for (int s = 128; s > 0; s >>= 1) {
        if (tid < s) { s1[tid] += s1[tid + s]; s2[tid] += s2[tid + s]; }
        __syncthreads();
    }
    const float mu  = s1[0] * (1.0f / HDIM);
    const float var = s2[0] * (1.0f / HDIM) - mu * mu;
    const float rs  = rsqrtf(var + LNEPS);
    float* orow = out + (size_t)row * HDIM;
    for (int i = tid; i < HDIM; i += 256)
        orow[i] = (xr[i] - mu) * rs * gamma[i] + beta[i];
}

// ---------------------------------------------------------------------------
// 2) mark history token ids (set semantics -> single penalty application)
// ---------------------------------------------------------------------------
__global__ void __launch_bounds__(256) mark_hist_kernel(
    const int* __restrict__ ids, unsigned char* __restrict__ marks) {
    int i = blockIdx.x * 256 + threadIdx.x;      // i in [0, BATCH*HIST)
    int b = i / HIST;
    int id = ids[i];
    marks[(size_t)b * VOCAB + id] = 1;
}

// ---------------------------------------------------------------------------
// 3) fp32 GEMM via V_WMMA_F32_16X16X4_F32, penalty fused in epilogue
//    logits[b][v] = sum_h hnorm[b][h] * W[v][h]
//    Block: 128 threads = 4 waves; each wave: M=32 x N=64 (4 N-tiles)
// ---------------------------------------------------------------------------
__global__ void __launch_bounds__(128) gemm_wmma_kernel(
    const float* __restrict__ h, const float* __restrict__ W,
    const unsigned char* __restrict__ marks, float* __restrict__ logits) {
    const int lane  = threadIdx.x & 31;
    const int wave  = threadIdx.x >> 5;
    const int nbase = blockIdx.x * 256 + wave * 64;
    const int lo    = lane & 15;          // column / row-in-tile index
    const int hi    = lane >> 4;          // 0: K pair {0,1}; 1: K pair {2,3}
    const int koff  = hi * 2;

    v8f acc[2][4] = {};                    // [M-tile][N-tile]
    const float* hrow0 = h + (size_t)lo * HDIM;          // M = lo
    const float* hrow1 = h + (size_t)(lo + 16) * HDIM;   // M = lo + 16

    for (int k = 0; k < HDIM; k += 4) {
        // A operands: 16x4 fp32 tiles (lanes 0-15 K={k,k+1}, lanes 16-31 K={k+2,k+3})
        v2f a0 = *(const v2f*)(hrow0 + k + koff);
        v2f a1 = *(const v2f*)(hrow1 + k + koff);
#pragma unroll
        for (int t = 0; t < 4; ++t) {
            const int col = nbase + t * 16 + lo;
            // B operand: 4x16 tile of W^T; contiguous float2 along W's row
            v2f bb = *(const v2f*)(W + (size_t)col * HDIM + k + koff);
            acc[0][t] = __builtin_amdgcn_wmma_f32_16x16x4_f32(
                false, a0, false, bb, (short)0, acc[0][t], false, false);
            acc[1][t] = __builtin_amdgcn_wmma_f32_16x16x4_f32(
                false, a1, false, bb, (short)0, acc[1][t], false, false);
        }
    }

    // Epilogue: C/D layout -> VGPR r, lanes 0-15: M=r, lanes 16-31: M=r+8
#pragma unroll
    for (int m = 0; m < 2; ++m) {
#pragma unroll
        for (int t = 0; t < 4; ++t) {
            const int n = nbase + t * 16 + lo;
#pragma unroll
            for (int r = 0; r < 8; ++r) {
                const int M = m * 16 + hi * 8 + r;
                float v = acc[m][t][r];
                if (marks[(size_t)M * VOCAB + n])
                    v = (v < 0.0f) ? v * PEN : v / PEN;
                logits[(size_t)M * VOCAB + n] = v;
            }
        }
    }
}

// ---------------------------------------------------------------------------
// 4) top-50 per row: 50 rounds of block-wide argmax (lowest-index tiebreak)
// ---------------------------------------------------------------------------
__global__ void __launch_bounds__(1024) topk_kernel(
    float* __restrict__ logits, float* __restrict__ tvals, int* __restrict__ tidx) {
    __shared__ float sv[1024];
    __shared__ int   si[1024];
    const int b = blockIdx.x, tid = threadIdx.x;
    float* row = logits + (size_t)b * VOCAB;
    for (int r = 0; r < TOPK; ++r) {
        float best = -INFINITY; int bi = 0x7fffffff;
        for (int i = tid; i < VOCAB; i += 1024) {
            float v = row[i];
            if (v > best) { best = v; bi = i; }   // ascending i -> lowest index kept
        }
        sv[tid] = best; si[tid] = bi;
        __syncthreads();
        for (int s = 512; s > 0; s >>= 1) {
            if (tid < s) {
                float ov = sv[tid + s]; int oi = si[tid + s];
                if (ov > sv[tid] || (ov == sv[tid] && oi < si[tid])) {
                    sv[tid] = ov; si[tid] = oi;
                }
            }
            __syncthreads();
        }
        if (tid == 0) {
            tvals[b * TOPK + r] = sv[0];
            tidx [b * TOPK + r] = si[0];
            row[si[0]] = -INFINITY;               // mask out for next round
        }
        __syncthreads();
    }
}

// ---------------------------------------------------------------------------
// 5) temperature / softmax / cumsum / top-p mask / softmax  (50 elems per row)
//    out layout: probs [32*50] floats, then tokens [32*50] (value-cast)
// ---------------------------------------------------------------------------
__global__ void __launch_bounds__(32) finalize_kernel(
    const float* __restrict__ tvals, const int* __restrict__ tidx,
    float* __restrict__ out) {
    if (threadIdx.x != 0) return;
    const int b = blockIdx.x;
    float v[TOPK], p[TOPK], f[TOPK];
    for (int i = 0; i < TOPK; ++i) v[i] = tvals[b * TOPK + i];  // TEMPERATURE == 1.0
    // softmax (vals sorted descending -> max is v[0])
    float s = 0.f;
    for (int i = 0; i < TOPK; ++i) { p[i] = expf(v[i] - v[0]); s += p[i]; }
    const float inv = 1.0f / s;
    // inclusive cumsum + nucleus mask
    float cum = 0.f;
    for (int i = 0; i < TOPK; ++i) {
        cum += p[i] * inv;
        bool keep = (cum < TOPP) || (i < MINKEEP);
        f[i] = keep ? v[i] : -1000.0f;
    }
    // second softmax (f[0] == v[0] is the max: always kept by MINKEEP)
    float s2 = 0.f;
    for (int i = 0; i < TOPK; ++i) { f[i] = expf(f[i] - v[0]); s2 += f[i]; }
    const float inv2 = 1.0f / s2;
    for (int i = 0; i < TOPK; ++i) {
        out[b * TOPK + i] = f[i] * inv2;
        out[BATCH * TOPK + b * TOPK + i] = (float)tidx[b * TOPK + i];
    }
}

// ---------------------------------------------------------------------------
// launch
// ---------------------------------------------------------------------------
extern "C" void kernel_launch(void* const* d_in, const int* in_sizes, int n_in,
                              void* d_out, int out_size, void* d_ws, size_t ws_size,
                              hipStream_t stream) {
    const int*   input_ids = (const int*)  d_in[0];
    const float* hidden    = (const float*)d_in[1];
    const float* gamma     = (const float*)d_in[2];
    const float* beta      = (const float*)d_in[3];
    const float* W         = (const float*)d_in[4];
    float* out = (float*)d_out;

    // workspace layout (bytes)
    char* ws = (char*)d_ws;
    float*         h_norm = (float*)(ws);                              // 256 KB
    float*         logits = (float*)(ws + 262144);                     // 16 MB
    unsigned char* marks  = (unsigned char*)(ws + 262144 + 16384000);  // 4 MB
    float*         tvals  = (float*)(ws + 262144 + 16384000 + 4096000);
    int*           tidx   = (int*)  (ws + 262144 + 16384000 + 4096000 + 6400);

    const int mark16 = (BATCH * VOCAB) / 16;  // 256000 uint4
    zero_marks_kernel<<<(mark16 + 255) / 256, 256, 0, stream>>>((uint4*)marks, mark16);
    layernorm_kernel<<<BATCH, 256, 0, stream>>>(hidden, gamma, beta, h_norm);
    mark_hist_kernel<<<(BATCH * HIST) / 256, 256, 0, stream>>>(input_ids, marks);
    gemm_wmma_kernel<<<VOCAB / 256, 128, 0, stream>>>(h_norm, W, marks, logits);
    topk_kernel<<<BATCH, 1024, 0, stream>>>(logits, tvals, tidx);
    finalize_kernel<<<BATCH, 32, 0, stream>>>(tvals, tidx, out);
}